// GCN_57724360458897
// MI455X (gfx1250) — compile-verified
//
#include <hip/hip_runtime.h>
#include <hip/hip_bf16.h>
#include <math.h>

#define NNODES 100000
#define NEDGES 1600000
#define EMB    128
#define HID    256
#define NCLS   64

typedef float v2f __attribute__((ext_vector_type(2)));
typedef float v8f __attribute__((ext_vector_type(8)));

__device__ __forceinline__ float gelu_exact(float x) {
    // exact erf GELU (matches torch nn.GELU / jax approximate=False)
    return 0.5f * x * (1.0f + erff(x * 0.70710678118654752f));
}

// ---------------------------------------------------------------------------
// Zero workspace (agg buffers)
// ---------------------------------------------------------------------------
__global__ void zero_kernel(float* __restrict__ p, int n) {
    int i = blockIdx.x * blockDim.x + threadIdx.x;
    int stride = gridDim.x * blockDim.x;
    for (; i < n; i += stride) p[i] = 0.0f;
}

// ---------------------------------------------------------------------------
// Edge scatter-add: agg[dst[e]] += (RELU ? max(X[src[e]],0) : X[src[e]])
// One wave32 per edge; float4 gathers + global_atomic_add_f32.
// ---------------------------------------------------------------------------
template<int F, bool RELU>
__global__ void scatter_add_kernel(const float* __restrict__ X,
                                   const int* __restrict__ src,
                                   const int* __restrict__ dst,
                                   float* __restrict__ agg, int nEdges) {
    int wave = threadIdx.x >> 5;           // 8 waves / 256-thread block
    int lane = threadIdx.x & 31;
    int e = blockIdx.x * 8 + wave;
    if (e >= nEdges) return;
    int s = src[e];
    int d = dst[e];
    const float* xrow = X + (size_t)s * F;
    float* arow = agg + (size_t)d * F;
    __builtin_prefetch(arow, 1, 1);        // global_prefetch_b8 on gfx1250
    constexpr int CH = F / 128;            // float4 chunks per lane (1 or 2)
#pragma unroll
    for (int c = 0; c < CH; ++c) {
        int j = (c * 32 + lane) * 4;
        float4 v = *(const float4*)(xrow + j);
        if (RELU) {
            v.x = fmaxf(v.x, 0.0f);
            v.y = fmaxf(v.y, 0.0f);
            v.z = fmaxf(v.z, 0.0f);
            v.w = fmaxf(v.w, 0.0f);
        }
        atomicAdd(arow + j + 0, v.x);
        atomicAdd(arow + j + 1, v.y);
        atomicAdd(arow + j + 2, v.z);
        atomicAdd(arow + j + 3, v.w);
    }
}

// ---------------------------------------------------------------------------
// Fused A-fragment load: float2 at (row base + k + 2*half)
//   IMODE 0: Xa      IMODE 1: Xa + Xb      IMODE 2: relu(Xa) + Xb
// ---------------------------------------------------------------------------
template<int IMODE>
__device__ __forceinline__ float2 load_a_frag(const float* __restrict__ xa,
                                              const float* __restrict__ xb,
                                              int off) {
    float2 v = *(const float2*)(xa + off);
    if (IMODE == 1) {
        float2 w = *(const float2*)(xb + off);
        v.x += w.x; v.y += w.y;
    } else if (IMODE == 2) {
        float2 w = *(const float2*)(xb + off);
        v.x = fmaxf(v.x, 0.0f) + w.x;
        v.y = fmaxf(v.y, 0.0f) + w.y;
    }
    return v;
}

// ---------------------------------------------------------------------------
// fp32 WMMA GEMM: Out[M,Nout] = act( X[M,K] @ W[K,Nout] + bias )
// Block = 128 threads (4 waves). blockIdx.y selects a 64-col group; the
// K x 64 weight block is staged once into LDS, pair-swizzled so one B
// fragment = one aligned ds_load_b64. Each wave computes a 32x64 strip
// (2 row tiles x 4 col tiles = 8 accumulators), so every B fragment feeds
// two v_wmma_f32_16x16x4_f32. A-fragment loads are software-pipelined.
// ---------------------------------------------------------------------------
template<int K, int IMODE, int ACT>
__global__ __launch_bounds__(128)
void gemm_wmma_kernel(const float* __restrict__ Xa,
                      const float* __restrict__ Xb,
                      const float* __restrict__ W,
                      const float* __restrict__ bias,
                      float* __restrict__ Out,
                      int Nout, int npairs) {
    // pair-interleaved layout: wlds[((k>>1)*64 + n)*2 + (k&1)] = W[k][col0+n]
    __shared__ float wlds[K * 64];

    const int tid  = threadIdx.x;
    const int col0 = blockIdx.y * 64;

    // ---- stage W block [K x 64] into LDS (all 128 threads) ----
#pragma unroll 4
    for (int idx = tid; idx < K * 64; idx += 128) {
        int kk = idx >> 6;                 // 0..K-1 (row)
        int n  = idx & 63;                 // 0..63  (col) -> coalesced reads
        float w = W[(size_t)kk * Nout + (col0 + n)];
        wlds[(((kk >> 1) * 64) + n) * 2 + (kk & 1)] = w;
    }
    __syncthreads();

    const int wave = tid >> 5;
    const int lane = tid & 31;
    const int pairIdx = blockIdx.x * 4 + wave;   // 32-row strip index
    if (pairIdx >= npairs) return;               // wave-uniform: EXEC all-1s
    const int half = lane >> 4;                  // lane half
    const int m    = lane & 15;

    const int rowA = pairIdx * 32 + m;           // A rows: tile0, tile1 = +16
    const float* xa0 = Xa + (size_t)rowA * K + 2 * half;
    const float* xa1 = xa0 + (size_t)16 * K;
    const float* xb0 = (IMODE != 0) ? (Xb + (size_t)rowA * K + 2 * half) : xa0;
    const float* xb1 = (IMODE != 0) ? (xb0 + (size_t)16 * K) : xa1;

    v8f acc[2][4];
#pragma unroll
    for (int rt = 0; rt < 2; ++rt)
#pragma unroll
        for (int t = 0; t < 4; ++t)
#pragma unroll
            for (int r = 0; r < 8; ++r) acc[rt][t][r] = 0.0f;

    const float2* wp = (const float2*)wlds;      // [K/2][64] k-pairs

    // ---- software-pipelined K loop, 8 WMMAs per step ----
    float2 a0 = load_a_frag<IMODE>(xa0, xb0, 0);
    float2 a1 = load_a_frag<IMODE>(xa1, xb1, 0);
#pragma unroll 2
    for (int k = 0; k < K; k += 4) {
        v2f A0; A0.x = a0.x; A0.y = a0.y;
        v2f A1; A1.x = a1.x; A1.y = a1.y;
        if (k + 4 < K) {                         // prefetch next fragments
            a0 = load_a_frag<IMODE>(xa0, xb0, k + 4);
            a1 = load_a_frag<IMODE>(xa1, xb1, k + 4);
        }
        const float2* prow = wp + ((k >> 1) + half) * 64 + m;
#pragma unroll
        for (int t = 0; t < 4; ++t) {
            float2 bb = prow[t * 16];            // ds_load_b64, imm offset
            v2f B; B.x = bb.x; B.y = bb.y;
            acc[0][t] = __builtin_amdgcn_wmma_f32_16x16x4_f32(
                false, A0, false, B, (short)0, acc[0][t], false, false);
            acc[1][t] = __builtin_amdgcn_wmma_f32_16x16x4_f32(
                false, A1, false, B, (short)0, acc[1][t], false, false);
        }
    }

    // ---- epilogue: bias + activation + store (documented C/D layout) ----
    float bvals[4];
#pragma unroll
    for (int t = 0; t < 4; ++t) bvals[t] = bias[col0 + t * 16 + m];
#pragma unroll
    for (int rt = 0; rt < 2; ++rt) {
        int rbase = pairIdx * 32 + rt * 16 + 8 * half;
#pragma unroll
        for (int r = 0; r < 8; ++r) {
            float* op = Out + (size_t)(rbase + r) * Nout + col0;
#pragma unroll
            for (int t = 0; t < 4; ++t) {
                float v = acc[rt][t][r] + bvals[t];
                if (ACT == 1) v = gelu_exact(v);
                op[t * 16 + m] = v;
            }
        }
    }
}

// ---------------------------------------------------------------------------
// Orchestration
// ---------------------------------------------------------------------------
extern "C" void kernel_launch(void* const* d_in, const int* in_sizes, int n_in,
                              void* d_out, int out_size, void* d_ws, size_t ws_size,
                              hipStream_t stream) {
    const float* emb = (const float*)d_in[0];
    const int*   src = (const int*)d_in[1];
    const int*   dst = (const int*)d_in[2];
    const float* w1a = (const float*)d_in[3];
    const float* b1a = (const float*)d_in[4];
    const float* w2a = (const float*)d_in[5];
    const float* b2a = (const float*)d_in[6];
    const float* w1b = (const float*)d_in[7];
    const float* b1b = (const float*)d_in[8];
    const float* w2b = (const float*)d_in[9];
    const float* b2b = (const float*)d_in[10];

    float* out_h  = (float*)d_out;                       // [N, 64]
    float* out_h0 = out_h + (size_t)NNODES * NCLS;       // [N, 256] (pre-relu h0)

    float* agg1 = (float*)d_ws;                          // [N, 128]
    float* agg2 = agg1 + (size_t)NNODES * EMB;           // [N, 256]
    float* hmid = agg2 + (size_t)NNODES * HID;           // [N, 256] scratch hidden

    // 1) zero both agg buffers (contiguous)
    zero_kernel<<<2048, 256, 0, stream>>>(agg1, NNODES * (EMB + HID));

    const int sgrid  = (NEDGES + 7) / 8;                 // 8 edges / 256-thr block
    const int npairs = NNODES / 32;                      // 3125, exact
    dim3 gblk(128);
    dim3 gHID((npairs + 3) / 4, HID / 64);
    dim3 gCLS((npairs + 3) / 4, NCLS / 64);

    // 2) layer-1 aggregation: agg1[d] += emb[s]
    scatter_add_kernel<EMB, false><<<sgrid, 256, 0, stream>>>(emb, src, dst, agg1, NEDGES);

    // 3) hmid = gelu( (emb + agg1) @ w1a + b1a )
    gemm_wmma_kernel<EMB, 1, 1><<<gHID, gblk, 0, stream>>>(emb, agg1, w1a, b1a, hmid, HID, npairs);

    // 4) h0 = hmid @ w2a + b2a   (output #2)
    gemm_wmma_kernel<HID, 0, 0><<<gHID, gblk, 0, stream>>>(hmid, nullptr, w2a, b2a, out_h0, HID, npairs);

    // 5) layer-2 aggregation: agg2[d] += relu(h0[s])
    scatter_add_kernel<HID, true><<<sgrid, 256, 0, stream>>>(out_h0, src, dst, agg2, NEDGES);

    // 6) hmid = gelu( (relu(h0) + agg2) @ w1b + b1b )
    gemm_wmma_kernel<HID, 2, 1><<<gHID, gblk, 0, stream>>>(out_h0, agg2, w1b, b1b, hmid, HID, npairs);

    // 7) h = hmid @ w2b + b2b    (output #1)
    gemm_wmma_kernel<HID, 0, 0><<<gCLS, gblk, 0, stream>>>(hmid, nullptr, w2b, b2b, out_h, NCLS, npairs);
}